// NGCFModel_47888885350522
// MI455X (gfx1250) — compile-verified
//
#include <hip/hip_runtime.h>
#include <hip/hip_bf16.h>

typedef float v2f __attribute__((ext_vector_type(2)));
typedef float v4f __attribute__((ext_vector_type(4)));
typedef float v8f __attribute__((ext_vector_type(8)));
typedef int   v4i __attribute__((ext_vector_type(4)));

#define NUM_USERS 100000
#define NUM_ITEMS 50000
#define N_NODES   (NUM_USERS + NUM_ITEMS)   // 150000
#define EMBED_K   64
#define N_LAYERS  3
#define N_EDGES   2000000
#define BATCH     4096
#define ROWSTRIDE (EMBED_K * (N_LAYERS + 1)) // 256 floats per node row
#define NEG_SLOPE 0.01f
#define EPS_NRM   1e-12f

// All indices into the 153.6MB embedding table fit in 32 bits -> use unsigned
// offsets everywhere so the compiler emits saddr + 32-bit voffset addressing
// instead of 64-bit pointer arithmetic chains.

// ---------------------------------------------------------------------------
// K0: write x0 = concat(Gu, Gi) into cols [0,64) of the big emb table
// ---------------------------------------------------------------------------
__global__ void init_x0_kernel(const float* __restrict__ Gu,
                               const float* __restrict__ Gi,
                               float* __restrict__ embs) {
    unsigned idx = blockIdx.x * blockDim.x + threadIdx.x;   // over N_NODES*64
    if (idx >= N_NODES * EMBED_K) return;
    unsigned node = idx >> 6;
    unsigned c    = idx & 63u;
    float v = (node < NUM_USERS) ? Gu[idx] : Gi[idx - NUM_USERS * EMBED_K];
    embs[node * ROWSTRIDE + c] = v;
}

// ---------------------------------------------------------------------------
// K1: self transform. For 16-node tiles: out = x_tile @ W1[l] + b1[l],
//     written to cols [col_next, col_next+64). fp32 WMMA 16x16x4, K-loop of 16.
// ---------------------------------------------------------------------------
__global__ __launch_bounds__(256)
void self_transform_kernel(float* __restrict__ embs, int col_l, int col_next,
                           const float* __restrict__ Wl,
                           const float* __restrict__ bl, int n_tiles) {
    // Swizzled B fragments: WB[kk][nt][lane][2], kk=0..15, nt=0..3
    __shared__ float WB[16 * 4 * 32 * 2];   // 16 KB
    for (int idx = threadIdx.x; idx < 16 * 4 * 32; idx += blockDim.x) {
        int kk   = idx >> 7;
        int rem  = idx & 127;
        int nt   = rem >> 5;
        int lane = rem & 31;
        int half = lane >> 4, l16 = lane & 15;
        int n  = nt * 16 + l16;
        int r0 = 4 * kk + 2 * half;
        WB[idx * 2 + 0] = Wl[r0 * 64 + n];
        WB[idx * 2 + 1] = Wl[(r0 + 1) * 64 + n];
    }
    __syncthreads();

    int lane = threadIdx.x & 31;
    int half = lane >> 4, l16 = lane & 15;
    int wave   = blockIdx.x * (blockDim.x >> 5) + (threadIdx.x >> 5);
    int nwaves = gridDim.x * (blockDim.x >> 5);

    for (int tile = wave; tile < n_tiles; tile += nwaves) {
        unsigned xbase = (unsigned)(tile * 16 + l16) * ROWSTRIDE + col_l;
        v2f a[16];
#pragma unroll
        for (int kk = 0; kk < 16; ++kk)
            a[kk] = *(const v2f*)&embs[xbase + 4 * kk + 2 * half];

#pragma unroll
        for (int nt = 0; nt < 4; ++nt) {
            float bv = bl[nt * 16 + l16];
            v8f acc = {bv, bv, bv, bv, bv, bv, bv, bv};
#pragma unroll
            for (int kk = 0; kk < 16; ++kk) {
                v2f bfrag = *(const v2f*)&WB[((kk * 4 + nt) * 32 + lane) * 2];
                acc = __builtin_amdgcn_wmma_f32_16x16x4_f32(
                    false, a[kk], false, bfrag, (short)0, acc, false, false);
            }
#pragma unroll
            for (int v = 0; v < 8; ++v) {
                int m = v + 8 * half;
                embs[(unsigned)(tile * 16 + m) * ROWSTRIDE + col_next +
                     nt * 16 + l16] = acc[v];
            }
        }
    }
}

// ---------------------------------------------------------------------------
// K2: edge messages. msg = [xj | xj*xi] (16x128) @ [W1;W2] (128x64) + (b1+b2),
//     scaled by edge_w, atomically scattered into embs[:, col_next).
// ---------------------------------------------------------------------------
__global__ __launch_bounds__(256)
void edge_msg_kernel(float* __restrict__ embs, int col_l, int col_next,
                     const float* __restrict__ W1l, const float* __restrict__ b1l,
                     const float* __restrict__ W2l, const float* __restrict__ b2l,
                     const float* __restrict__ edge_w,
                     const int* __restrict__ edge_src,
                     const int* __restrict__ edge_dst, int n_tiles) {
    // Swizzled B fragments for combined K=128: kk=0..31 (0..15 -> W1, 16..31 -> W2)
    __shared__ float WB[32 * 4 * 32 * 2];   // 32 KB
    for (int idx = threadIdx.x; idx < 32 * 4 * 32; idx += blockDim.x) {
        int kk   = idx >> 7;
        int rem  = idx & 127;
        int nt   = rem >> 5;
        int lane = rem & 31;
        int half = lane >> 4, l16 = lane & 15;
        int n  = nt * 16 + l16;
        const float* Ws = (kk < 16) ? W1l : W2l;
        int r0 = 4 * (kk & 15) + 2 * half;
        WB[idx * 2 + 0] = Ws[r0 * 64 + n];
        WB[idx * 2 + 1] = Ws[(r0 + 1) * 64 + n];
    }
    __syncthreads();

    int lane = threadIdx.x & 31;
    int half = lane >> 4, l16 = lane & 15;
    int wave   = blockIdx.x * (blockDim.x >> 5) + (threadIdx.x >> 5);
    int nwaves = gridDim.x * (blockDim.x >> 5);

    for (int tile = wave; tile < n_tiles; tile += nwaves) {
        int e  = tile * 16 + l16;       // this lane's A-row edge
        unsigned sj = (unsigned)edge_src[e];
        unsigned di = (unsigned)edge_dst[e];
        unsigned xjb = sj * ROWSTRIDE + col_l;
        unsigned xib = di * ROWSTRIDE + col_l;

        v2f a[32];
#pragma unroll
        for (int kk = 0; kk < 16; ++kk) {
            v2f aj = *(const v2f*)&embs[xjb + 4 * kk + 2 * half];
            v2f ai = *(const v2f*)&embs[xib + 4 * kk + 2 * half];
            a[kk]      = aj;
            a[kk + 16] = aj * ai;
        }

        // edge weight / destination for the 8 output rows this lane owns
        // (32B-aligned vector loads: base = tile*16 + half*8)
        int ebase = tile * 16 + half * 8;
        v4f ew0 = *(const v4f*)&edge_w[ebase];
        v4f ew1 = *(const v4f*)&edge_w[ebase + 4];
        v4i dn0 = *(const v4i*)&edge_dst[ebase];
        v4i dn1 = *(const v4i*)&edge_dst[ebase + 4];
        float ew[8] = {ew0.x, ew0.y, ew0.z, ew0.w, ew1.x, ew1.y, ew1.z, ew1.w};
        unsigned dn[8];
        dn[0] = (unsigned)dn0.x; dn[1] = (unsigned)dn0.y;
        dn[2] = (unsigned)dn0.z; dn[3] = (unsigned)dn0.w;
        dn[4] = (unsigned)dn1.x; dn[5] = (unsigned)dn1.y;
        dn[6] = (unsigned)dn1.z; dn[7] = (unsigned)dn1.w;

#pragma unroll
        for (int nt = 0; nt < 4; ++nt) {
            int n = nt * 16 + l16;
            float bv = b1l[n] + b2l[n];
            v8f acc = {bv, bv, bv, bv, bv, bv, bv, bv};
#pragma unroll
            for (int kk = 0; kk < 32; ++kk) {
                v2f bfrag = *(const v2f*)&WB[((kk * 4 + nt) * 32 + lane) * 2];
                acc = __builtin_amdgcn_wmma_f32_16x16x4_f32(
                    false, a[kk], false, bfrag, (short)0, acc, false, false);
            }
#pragma unroll
            for (int v = 0; v < 8; ++v) {
                atomicAdd(&embs[dn[v] * ROWSTRIDE + (unsigned)(col_next + n)],
                          acc[v] * ew[v]);
            }
        }
    }
}

// ---------------------------------------------------------------------------
// K3: LeakyReLU + row L2 normalize in place on cols [col, col+64). Wave/row.
// ---------------------------------------------------------------------------
__global__ __launch_bounds__(256)
void act_norm_kernel(float* __restrict__ embs, int col, int n_rows) {
    int wave = blockIdx.x * (blockDim.x >> 5) + (threadIdx.x >> 5);
    int lane = threadIdx.x & 31;
    if (wave >= n_rows) return;
    unsigned base = (unsigned)wave * ROWSTRIDE + col + lane * 2;
    v2f v = *(v2f*)&embs[base];
    v2f o;
    o.x = (v.x > 0.f) ? v.x : NEG_SLOPE * v.x;
    o.y = (v.y > 0.f) ? v.y : NEG_SLOPE * v.y;
    float s = o.x * o.x + o.y * o.y;
#pragma unroll
    for (int off = 16; off > 0; off >>= 1) s += __shfl_xor(s, off, 32);
    float scale = 1.0f / fmaxf(sqrtf(s), EPS_NRM);
    o.x *= scale;
    o.y *= scale;
    *(v2f*)&embs[base] = o;
}

// ---------------------------------------------------------------------------
// K4: scores. One wave per batch element: 256-dim dot of user/item rows.
// ---------------------------------------------------------------------------
__global__ __launch_bounds__(256)
void score_kernel(const float* __restrict__ embs, const int* __restrict__ user,
                  const int* __restrict__ pos, float* __restrict__ out,
                  int batch) {
    int wave = blockIdx.x * (blockDim.x >> 5) + (threadIdx.x >> 5);
    int lane = threadIdx.x & 31;
    if (wave >= batch) return;
    unsigned u  = (unsigned)user[wave];
    unsigned it = (unsigned)pos[wave];
    unsigned ub = u * ROWSTRIDE + lane * 8;
    unsigned ib = (NUM_USERS + it) * ROWSTRIDE + lane * 8;
    v4f a0  = *(const v4f*)&embs[ub];
    v4f a1  = *(const v4f*)&embs[ub + 4];
    v4f b0  = *(const v4f*)&embs[ib];
    v4f b1v = *(const v4f*)&embs[ib + 4];
    v4f p = a0 * b0 + a1 * b1v;
    float s = p.x + p.y + p.z + p.w;
#pragma unroll
    for (int off = 16; off > 0; off >>= 1) s += __shfl_xor(s, off, 32);
    if (lane == 0) out[wave] = s;
}

// ---------------------------------------------------------------------------
extern "C" void kernel_launch(void* const* d_in, const int* in_sizes, int n_in,
                              void* d_out, int out_size, void* d_ws,
                              size_t ws_size, hipStream_t stream) {
    const float* Gu       = (const float*)d_in[0];
    const float* Gi       = (const float*)d_in[1];
    const float* W1       = (const float*)d_in[2];   // [3,64,64]
    const float* b1       = (const float*)d_in[3];   // [3,64]
    const float* W2       = (const float*)d_in[4];   // [3,64,64]
    const float* b2       = (const float*)d_in[5];   // [3,64]
    const float* edge_w   = (const float*)d_in[6];   // [E]
    const int*   edge_src = (const int*)d_in[7];     // [E]
    const int*   edge_dst = (const int*)d_in[8];     // [E]
    const int*   user     = (const int*)d_in[9];     // [B]
    const int*   pos      = (const int*)d_in[10];    // [B]
    float*       out      = (float*)d_out;           // [B]

    // Workspace: [N_NODES, 256] fp32 concat-embedding table (153.6 MB)
    float* embs = (float*)d_ws;

    const int node_tiles = N_NODES / 16;   // 9375
    const int edge_tiles = N_EDGES / 16;   // 125000

    {
        int total = N_NODES * EMBED_K;
        init_x0_kernel<<<(total + 255) / 256, 256, 0, stream>>>(Gu, Gi, embs);
    }

    for (int l = 0; l < N_LAYERS; ++l) {
        int col_l    = EMBED_K * l;
        int col_next = EMBED_K * (l + 1);
        const float* W1l = W1 + (size_t)l * 64 * 64;
        const float* b1l = b1 + (size_t)l * 64;
        const float* W2l = W2 + (size_t)l * 64 * 64;
        const float* b2l = b2 + (size_t)l * 64;

        // init next-layer cols with self transform (x @ W1 + b1)
        self_transform_kernel<<<1024, 256, 0, stream>>>(
            embs, col_l, col_next, W1l, b1l, node_tiles);
        // scatter edge messages on top (segment_sum via fp32 atomics)
        edge_msg_kernel<<<2048, 256, 0, stream>>>(
            embs, col_l, col_next, W1l, b1l, W2l, b2l, edge_w, edge_src,
            edge_dst, edge_tiles);
        // LeakyReLU + L2 normalize rows in place
        act_norm_kernel<<<(N_NODES + 7) / 8, 256, 0, stream>>>(
            embs, col_next, N_NODES);
    }

    score_kernel<<<(BATCH + 7) / 8, 256, 0, stream>>>(embs, user, pos, out,
                                                      BATCH);
}